// GCN_35416300322991
// MI455X (gfx1250) — compile-verified
//
#include <hip/hip_runtime.h>
#include <math.h>

typedef __attribute__((ext_vector_type(2))) float v2f;
typedef __attribute__((ext_vector_type(8))) float v8f;

#define D_IN  128
#define D_H   128
#define D_OUT 64

// ---------------------------------------------------------------- degree prep
__global__ void k_deg_init(float* __restrict__ deg, int n) {
    int i = blockIdx.x * blockDim.x + threadIdx.x;
    if (i < n) deg[i] = 1.0f;                      // self-loop
}

__global__ void k_deg_count(const int* __restrict__ dst, float* __restrict__ deg, int e) {
    int i = blockIdx.x * blockDim.x + threadIdx.x;
    if (i < e) atomicAdd(&deg[dst[i]], 1.0f);
}

__global__ void k_dinv(const float* __restrict__ deg, float* __restrict__ dinv, int n) {
    int i = blockIdx.x * blockDim.x + threadIdx.x;
    if (i < n) dinv[i] = rsqrtf(deg[i]);           // deg >= 1 always
}

// ------------------------------------------------- GEMM: Hs = (A @ W) * dinv
// One wave computes a 16-row x NCOL strip of C using V_WMMA_F32_16X16X4_F32.
// W is staged into LDS pair-interleaved: Wlds[(k/2)*NCOL + n] = {W[k][n], W[k+1][n]}
// so every B operand is one aligned ds_load_b64 straight into an even VGPR pair
// (no v_mov repacking in front of the WMMA).
// Epilogue scales rows by dinv and writes Hs (gather buffer) and agg (seeded
// with the self-loop contribution Hs[row]).
template <int NCOL>
__global__ __launch_bounds__(256) void k_gemm_scaled(
    const float* __restrict__ A, const float* __restrict__ W,
    const float* __restrict__ dinv, float* __restrict__ hs,
    float* __restrict__ agg, int M) {
    constexpr int K  = 128;
    constexpr int NT = NCOL / 16;

    __shared__ float Wlds[K * NCOL];   // viewed as (K/2, NCOL) float2
    for (int i = threadIdx.x; i < K * NCOL; i += 256) {
        const int k = i / NCOL;
        const int n = i - k * NCOL;
        Wlds[((k >> 1) * NCOL + n) * 2 + (k & 1)] = W[i];
    }
    __syncthreads();

    const int lane = threadIdx.x & 31;
    const int wave = threadIdx.x >> 5;
    const int m    = lane & 15;
    const int half = lane >> 4;
    const int row0 = (blockIdx.x * 8 + wave) * 16;
    if (row0 >= M) return;                         // wave-uniform, EXEC stays all-1

    v8f acc[NT] = {};
    const float* __restrict__ arow = A + (size_t)(row0 + m) * K + 2 * half;

    for (int k = 0; k < K; k += 4) {
        const float2 av = *(const float2*)(arow + k);
        v2f a; a[0] = av.x; a[1] = av.y;
        // pair index: half 0 covers k,k+1 ; half 1 covers k+2,k+3
        const float2* __restrict__ wp =
            (const float2*)Wlds + ((k >> 1) + half) * NCOL;
#pragma unroll
        for (int t = 0; t < NT; ++t) {
            const float2 bv = wp[t * 16 + m];
            v2f b; b[0] = bv.x; b[1] = bv.y;
            acc[t] = __builtin_amdgcn_wmma_f32_16x16x4_f32(
                false, a, false, b, (short)0, acc[t], false, false);
        }
    }

#pragma unroll
    for (int j = 0; j < 8; ++j) {
        const int r = row0 + j + 8 * half;
        const float dv = dinv[r];
        float* __restrict__ hp = hs  + (size_t)r * NCOL;
        float* __restrict__ ap = agg + (size_t)r * NCOL;
#pragma unroll
        for (int t = 0; t < NT; ++t) {
            const float v = acc[t][j] * dv;
            hp[t * 16 + m] = v;
            ap[t * 16 + m] = v;
        }
    }
}

// ---------------------------------------------- edge scatter: agg[dst] += Hs[src]
// One wave per edge; 128-bit coalesced gather of the source row (L2-resident),
// scalar f32 atomics into the destination row.
__global__ __launch_bounds__(256) void k_edges_d128(
    const int* __restrict__ src, const int* __restrict__ dst,
    const float* __restrict__ hs, float* __restrict__ agg, int e) {
    const int ei = blockIdx.x * 8 + (threadIdx.x >> 5);
    if (ei >= e) return;
    const int lane = threadIdx.x & 31;
    const int s = src[ei], d = dst[ei];
    const float4 v = ((const float4*)(hs + (size_t)s * 128))[lane];
    float* __restrict__ p = agg + (size_t)d * 128 + lane * 4;
    atomicAdd(p + 0, v.x);
    atomicAdd(p + 1, v.y);
    atomicAdd(p + 2, v.z);
    atomicAdd(p + 3, v.w);
}

__global__ __launch_bounds__(256) void k_edges_d64(
    const int* __restrict__ src, const int* __restrict__ dst,
    const float* __restrict__ hs, float* __restrict__ agg, int e) {
    const int ei = blockIdx.x * 8 + (threadIdx.x >> 5);
    if (ei >= e) return;
    const int lane = threadIdx.x & 31;
    const int s = src[ei], d = dst[ei];
    const float2 v = ((const float2*)(hs + (size_t)s * 64))[lane];
    float* __restrict__ p = agg + (size_t)d * 64 + lane * 2;
    atomicAdd(p + 0, v.x);
    atomicAdd(p + 1, v.y);
}

// ------------------------------------------- finalize layer1: dinv*agg + b, SELU
__global__ void k_finalize1(const float* __restrict__ agg, const float* __restrict__ dinv,
                            const float* __restrict__ b, float* __restrict__ out, int total) {
    int i = blockIdx.x * blockDim.x + threadIdx.x;
    if (i >= total) return;
    const int row = i >> 7;          // D_H = 128
    const int col = i & 127;
    float x = dinv[row] * agg[i] + b[col];
    const float alpha = 1.6732632423543772f;
    const float scale = 1.0507009873554805f;
    out[i] = scale * (x > 0.0f ? x : alpha * (__expf(x) - 1.0f));
}

// -------------------- finalize layer2: dinv*agg + b, then log_softmax over 64
// One wave per row; lane owns columns {lane, lane+32}; wave32 shuffle reductions.
__global__ __launch_bounds__(256) void k_finalize2(
    const float* __restrict__ agg, const float* __restrict__ dinv,
    const float* __restrict__ b, float* __restrict__ out, int n) {
    const int r = blockIdx.x * 8 + (threadIdx.x >> 5);
    if (r >= n) return;
    const int lane = threadIdx.x & 31;
    const float dv = dinv[r];
    float v0 = dv * agg[(size_t)r * 64 + lane]      + b[lane];
    float v1 = dv * agg[(size_t)r * 64 + lane + 32] + b[lane + 32];

    float mx = fmaxf(v0, v1);
#pragma unroll
    for (int off = 16; off >= 1; off >>= 1)
        mx = fmaxf(mx, __shfl_xor(mx, off, 32));

    float s = __expf(v0 - mx) + __expf(v1 - mx);
#pragma unroll
    for (int off = 16; off >= 1; off >>= 1)
        s += __shfl_xor(s, off, 32);

    const float l = mx + __logf(s);
    out[(size_t)r * 64 + lane]      = v0 - l;
    out[(size_t)r * 64 + lane + 32] = v1 - l;
}

// -------------------------------------------------------------------- launch
extern "C" void kernel_launch(void* const* d_in, const int* in_sizes, int n_in,
                              void* d_out, int out_size, void* d_ws, size_t ws_size,
                              hipStream_t stream) {
    const float* x    = (const float*)d_in[0];
    const int*   eidx = (const int*)  d_in[1];
    const float* W1   = (const float*)d_in[2];
    const float* b1   = (const float*)d_in[3];
    const float* W2   = (const float*)d_in[4];
    const float* b2   = (const float*)d_in[5];
    float* out = (float*)d_out;

    const int N = in_sizes[0] / D_IN;
    const int E = in_sizes[1] / 2;
    const int* src = eidx;          // edge_index[0]
    const int* dst = eidx + E;      // edge_index[1]

    // Workspace layout (floats): deg[N] | dinv[N] | buf1[N*128] | agg1[N*128]
    // Layer-2 buffers (N*64 each) reuse the agg1 region after finalize1.
    float* ws   = (float*)d_ws;
    float* deg  = ws;
    float* dinv = ws + N;
    float* buf1 = ws + 2 * (size_t)N;                 // Hs1, then A1 in-place
    float* agg1 = buf1 + (size_t)N * D_H;
    float* hs2  = agg1;                               // reuse (N*64)
    float* agg2 = agg1 + (size_t)N * D_OUT;           // reuse (N*64)

    const int strips  = (N + 15) / 16;
    const int gblocks = (strips + 7) / 8;

    // degree -> dinv
    k_deg_init <<<(N + 255) / 256, 256, 0, stream>>>(deg, N);
    k_deg_count<<<(E + 255) / 256, 256, 0, stream>>>(dst, deg, E);
    k_dinv     <<<(N + 255) / 256, 256, 0, stream>>>(deg, dinv, N);

    // layer 1
    k_gemm_scaled<D_H><<<gblocks, 256, 0, stream>>>(x, W1, dinv, buf1, agg1, N);
    k_edges_d128<<<(E + 7) / 8, 256, 0, stream>>>(src, dst, buf1, agg1, E);
    k_finalize1<<<((N * D_H) + 255) / 256, 256, 0, stream>>>(agg1, dinv, b1, buf1, N * D_H);

    // layer 2
    k_gemm_scaled<D_OUT><<<gblocks, 256, 0, stream>>>(buf1, W2, dinv, hs2, agg2, N);
    k_edges_d64<<<(E + 7) / 8, 256, 0, stream>>>(src, dst, hs2, agg2, E);
    k_finalize2<<<(N + 7) / 8, 256, 0, stream>>>(agg2, dinv, b2, out, N);
}